// QuantumNATEnhanced_65481071398016
// MI455X (gfx1250) — compile-verified
//
#include <hip/hip_runtime.h>
#include <hip/hip_bf16.h>
#include <math.h>

typedef __attribute__((ext_vector_type(2))) float v2f;
typedef __attribute__((ext_vector_type(8))) float v8f;

#define WIRES 4
#define DIM 16
#define BLK_ELEMS 128
#define THREADS 256
#define EPS_F 1e-5f

// ws float layout
#define WS_U 0        // 512 floats: [0..255] Bt_re[k][n] = Re(T[n][k]); [256..511] Bt_im
#define WS_STATS 512  // 8 floats: mu[4], rstd[4]
#define WS_BSUMS 1024 // nBlocks*8 floats: per-block {sum[4], sumsq[4]}

// ---------------------------------------------------------------------------
// Kernel A: build the 16x16 complex circuit unitary T, store T^T (re/im) in ws.
// Each of 16 lanes owns one column c of T (T e_c).
// ---------------------------------------------------------------------------
__global__ void build_unitary_kernel(const float* __restrict__ params,
                                     float* __restrict__ wsU) {
    int c = threadIdx.x;
    if (c >= DIM) return;
    float ur[DIM], ui[DIM];
#pragma unroll
    for (int i = 0; i < DIM; ++i) { ur[i] = (i == c) ? 1.f : 0.f; ui[i] = 0.f; }

#pragma unroll
    for (int l = 0; l < 2; ++l) {
        // single-qubit rotations on each wire
#pragma unroll
        for (int w = 0; w < WIRES; ++w) {
            float phi = params[l * 12 + w * 3 + 0];
            float th  = params[l * 12 + w * 3 + 1];
            float om  = params[l * 12 + w * 3 + 2];
            float cth = cosf(th * 0.5f), sth = sinf(th * 0.5f);
            float apo = 0.5f * (phi + om), amo = 0.5f * (phi - om);
            float epr = cosf(apo), epi = -sinf(apo);   // exp(-i(phi+om)/2)
            float emr = cosf(amo), emi =  sinf(amo);   // exp(+i(phi-om)/2)
            float m00r =  epr * cth, m00i =  epi * cth;
            float m01r = -emr * sth, m01i = -emi * sth;
            float m10r =  emr * sth, m10i = -emi * sth;
            float m11r =  epr * cth, m11i = -epi * cth;
            int bitw = 1 << (3 - w);
#pragma unroll
            for (int r0 = 0; r0 < DIM; ++r0) {
                if (r0 & bitw) continue;
                int r1 = r0 | bitw;
                float o0r = ur[r0], o0i = ui[r0];
                float o1r = ur[r1], o1i = ui[r1];
                ur[r0] = m00r * o0r - m00i * o0i + m01r * o1r - m01i * o1i;
                ui[r0] = m00r * o0i + m00i * o0r + m01r * o1i + m01i * o1r;
                ur[r1] = m10r * o0r - m10i * o0i + m11r * o1r - m11i * o1i;
                ui[r1] = m10r * o0i + m10i * o0r + m11r * o1i + m11i * o1r;
            }
        }
        // CNOT ring permutations: new[i] = old[perm[i]]
        int r = l % 3 + 1;
#pragma unroll
        for (int w = 0; w < WIRES; ++w) {
            int tq = (w + r) % WIRES;
            float tr[DIM], ti[DIM];
#pragma unroll
            for (int i = 0; i < DIM; ++i) {
                int perm = i ^ (((i >> (3 - w)) & 1) << (3 - tq));
                tr[i] = ur[perm]; ti[i] = ui[perm];
            }
#pragma unroll
            for (int i = 0; i < DIM; ++i) { ur[i] = tr[i]; ui[i] = ti[i]; }
        }
    }
    // store transposed for WMMA B operand: Bt[k=c][n] = T[n][c]
#pragma unroll
    for (int n = 0; n < DIM; ++n) {
        wsU[c * DIM + n]       = ur[n];
        wsU[256 + c * DIM + n] = ui[n];
    }
}

// ---------------------------------------------------------------------------
// Kernel B: pooling + WMMA batched matvec + per-wire expectations + partials.
// 256 threads (8 waves), 128 batch elements per block, 16 elements per wave.
// ---------------------------------------------------------------------------
__global__ __launch_bounds__(THREADS) void circuit_kernel(
    const float* __restrict__ x, const float* __restrict__ wsU,
    float* __restrict__ rawOut, float* __restrict__ blockSums) {
    __shared__ float sS[BLK_ELEMS][17];   // pooled states (pitch 17: conflict-free)
    __shared__ float sP[BLK_ELEMS][17];   // probabilities
    __shared__ float sInv[BLK_ELEMS];     // 1 / sum(s^2)
    __shared__ float sU[512];             // Bt_re | Bt_im
    __shared__ float sRed[4][8];          // per-wave partial reductions

    const int t   = threadIdx.x;
    const int blk = blockIdx.x;
    const long long base = (long long)blk * BLK_ELEMS;

    // preload T^T into LDS
    sU[t]       = wsU[t];
    sU[t + 256] = wsU[t + 256];

    // ---- phase 1: 6x6 average pooling (sum; scale cancels) ----
    const float* xb = x + base * 576;
#pragma unroll
    for (int v = 0; v < 8; ++v) {
        int gi = v * THREADS + t;            // 0..2047
        int e  = gi >> 4;                    // element in block
        int p  = gi & 15;                    // pooled index
        int pi = p >> 2, pj = p & 3;
        const float* bp = xb + e * 576 + pi * 144 + pj * 6;
        float sum = 0.f;
#pragma unroll
        for (int r = 0; r < 6; ++r) {
            const float2* row = (const float2*)(bp + r * 24);
            float2 a0 = row[0], a1 = row[1], a2 = row[2];
            sum += a0.x + a0.y + a1.x + a1.y + a2.x + a2.y;
        }
        sS[e][p] = sum;
    }
    __syncthreads();

    // ---- phase 2: 1/||s||^2 per element (consumed by same thread later) ----
    if (t < BLK_ELEMS) {
        float sq = 0.f;
#pragma unroll
        for (int k = 0; k < DIM; ++k) { float v = sS[t][k]; sq += v * v; }
        sInv[t] = 1.0f / sq;
    }

    // ---- phase 3: WMMA f32 16x16x4, chained over K=16, re & im ----
    const int lane = t & 31;
    const int e0   = (t >> 5) << 4;          // first element of this wave's tile
    const int mn   = lane & 15;              // M for A, N for B/D
    const int koff = (lane >> 4) << 1;       // 0 (lanes 0-15) or 2 (lanes 16-31)
    v8f cre = {}; v8f cim = {};
#pragma unroll
    for (int ck = 0; ck < 4; ++ck) {
        int k0 = ck * 4 + koff;
        v2f a  = { sS[e0 + mn][k0],            sS[e0 + mn][k0 + 1] };
        v2f br = { sU[k0 * 16 + mn],           sU[(k0 + 1) * 16 + mn] };
        v2f bi = { sU[256 + k0 * 16 + mn],     sU[256 + (k0 + 1) * 16 + mn] };
        cre = __builtin_amdgcn_wmma_f32_16x16x4_f32(false, a, false, br,
                                                    (short)0, cre, false, false);
        cim = __builtin_amdgcn_wmma_f32_16x16x4_f32(false, a, false, bi,
                                                    (short)0, cim, false, false);
    }
    // D layout: element row M = v + 8*(lane>=16), column N = lane&15
    const int hi = (lane >> 4) << 3;
#pragma unroll
    for (int v = 0; v < 8; ++v) {
        float re = cre[v], im = cim[v];
        sP[e0 + hi + v][mn] = re * re + im * im;
    }
    __syncthreads();

    // ---- phase 4: per-wire expectations + raw output ----
    float red[8];
    if (t < BLK_ELEMS) {
        float inv = sInv[t];
        float o0 = 0.f, o1 = 0.f, o2 = 0.f, o3 = 0.f;
#pragma unroll
        for (int nn = 0; nn < DIM; ++nn) {
            float p = sP[t][nn];
            o0 += (nn & 8) ? -p : p;   // wire 0 = bit 3
            o1 += (nn & 4) ? -p : p;
            o2 += (nn & 2) ? -p : p;
            o3 += (nn & 1) ? -p : p;
        }
        o0 *= inv; o1 *= inv; o2 *= inv; o3 *= inv;
        long long gb = (base + t) * 4;
        rawOut[gb + 0] = o0; rawOut[gb + 1] = o1;
        rawOut[gb + 2] = o2; rawOut[gb + 3] = o3;
        red[0] = o0; red[1] = o1; red[2] = o2; red[3] = o3;
        red[4] = o0 * o0; red[5] = o1 * o1; red[6] = o2 * o2; red[7] = o3 * o3;
        // deterministic in-wave tree reduction (wave32)
#pragma unroll
        for (int q = 0; q < 8; ++q) {
            float v = red[q];
            for (int off = 16; off > 0; off >>= 1) v += __shfl_xor(v, off, 32);
            red[q] = v;
        }
        if ((t & 31) == 0) {
#pragma unroll
            for (int q = 0; q < 8; ++q) sRed[t >> 5][q] = red[q];
        }
    }
    __syncthreads();
    if (t == 0) {
#pragma unroll
        for (int q = 0; q < 8; ++q)
            blockSums[blk * 8 + q] =
                sRed[0][q] + sRed[1][q] + sRed[2][q] + sRed[3][q];
    }
}

// ---------------------------------------------------------------------------
// Kernel C: deterministic reduction of block partials -> mu, rstd per wire.
// ---------------------------------------------------------------------------
__global__ void stats_kernel(const float* __restrict__ blockSums,
                             float* __restrict__ stats, int nBlocks, float invB) {
    __shared__ float part[8][32];
    __shared__ float tot[8];
    int t = threadIdx.x;          // 256 threads
    int q = t & 7, idx = t >> 3;  // idx 0..31
    float s = 0.f;
    for (int b = idx; b < nBlocks; b += 32) s += blockSums[b * 8 + q];
    part[q][idx] = s;
    __syncthreads();
    if (t < 8) {
        float sum = 0.f;
        for (int i = 0; i < 32; ++i) sum += part[t][i];
        tot[t] = sum * invB;
    }
    __syncthreads();
    if (t < 4) {
        float mu  = tot[t];
        float var = tot[t + 4] - mu * mu;
        stats[t]     = mu;
        stats[4 + t] = rsqrtf(var + EPS_F);
    }
}

// ---------------------------------------------------------------------------
// Kernel D: in-place batch-norm of raw outputs.
// ---------------------------------------------------------------------------
__global__ void norm_kernel(float* __restrict__ out,
                            const float* __restrict__ stats,
                            const float* __restrict__ gamma,
                            const float* __restrict__ beta, int n) {
    int i = blockIdx.x * blockDim.x + threadIdx.x;
    if (i >= n) return;
    int w = i & 3;
    out[i] = (out[i] - stats[w]) * stats[4 + w] * gamma[w] + beta[w];
}

extern "C" void kernel_launch(void* const* d_in, const int* in_sizes, int n_in,
                              void* d_out, int out_size, void* d_ws, size_t ws_size,
                              hipStream_t stream) {
    const float* x      = (const float*)d_in[0];
    const float* params = (const float*)d_in[1];
    const float* gamma  = (const float*)d_in[2];
    const float* beta   = (const float*)d_in[3];
    float* out = (float*)d_out;
    float* ws  = (float*)d_ws;

    int B = in_sizes[0] / 576;          // 65536
    int nBlocks = B / BLK_ELEMS;        // 512

    build_unitary_kernel<<<1, 32, 0, stream>>>(params, ws + WS_U);
    circuit_kernel<<<nBlocks, THREADS, 0, stream>>>(x, ws + WS_U, out,
                                                    ws + WS_BSUMS);
    stats_kernel<<<1, 256, 0, stream>>>(ws + WS_BSUMS, ws + WS_STATS, nBlocks,
                                        1.0f / (float)B);
    norm_kernel<<<(out_size + 255) / 256, 256, 0, stream>>>(out, ws + WS_STATS,
                                                            gamma, beta, out_size);
}